// edge_based_loss_75101798138063
// MI455X (gfx1250) — compile-verified
//
#include <hip/hip_runtime.h>
#include <hip/hip_bf16.h>
#include <stdint.h>

// Normalized 5-tap Gaussian, sigma=1: exp(-{4,1,0,1,4}/2)/sum
#define G0 0.05448868454481529f
#define G1 0.24420134200323332f
#define G2 0.40261994688390386f

#define IMG_W 512
#define IMG_H 512
#define HW    (IMG_W*IMG_H)
#define TILE  32
#define GROWS 38          // TILE + 2*3 halo rows
#define GCOLS 40          // TILE + 8 cols (gx in [ox0-4, ox0+36), 16B aligned)
#define N_PLANE (GROWS*GCOLS)   // 1520
#define N_GRAY  (GROWS*GROWS)   // 1444 (38x38, gx in [ox0-3, ox0+35))
#define HREG  34          // TILE + 2*1 halo for laplacian
#define N_HB   (GROWS*HREG)     // 1292
#define N_VB   (HREG*HREG)      // 1156

#if defined(__HIP_DEVICE_COMPILE__)

typedef float v2f __attribute__((ext_vector_type(2)));
typedef float v8f __attribute__((ext_vector_type(8)));

// gfx1250 async copies: global -> LDS, per-lane, tracked by ASYNCcnt.
// GVS mode: mem = SADDR(64) + VADDR(32); LDS dest byte offset in VDST vgpr.
__device__ __forceinline__ void async_ld_b32(uint32_t ldsOff, uint32_t vOff, uint64_t sBase) {
  asm volatile("global_load_async_to_lds_b32 %0, %1, %2"
               :: "v"(ldsOff), "v"(vOff), "s"(sBase) : "memory");
}
__device__ __forceinline__ void async_ld_b128(uint32_t ldsOff, uint32_t vOff, uint64_t sBase) {
  asm volatile("global_load_async_to_lds_b128 %0, %1, %2"
               :: "v"(ldsOff), "v"(vOff), "s"(sBase) : "memory");
}
__device__ __forceinline__ void wait_async0() {
  asm volatile("s_wait_asynccnt 0" ::: "memory");
}

// Wave32 sum via V_WMMA_F32_16X16X4_F32 with B = ones:
// a[0]=x, a[1]=0  ->  A[m][0]=x_m (lanes 0-15), A[m][2]=x_{m+16} (lanes 16-31)
// D[m][n] = x_m + x_{m+16}; per-lane sum of 8 D vgprs covers half the rows,
// shfl_xor(16) adds the other half -> full 32-lane sum in every lane.
__device__ __forceinline__ float wave_reduce_wmma(float x) {
  v2f a; a[0] = x;   a[1] = 0.0f;
  v2f b; b[0] = 1.0f; b[1] = 1.0f;
  v8f c = {0.f, 0.f, 0.f, 0.f, 0.f, 0.f, 0.f, 0.f};
  c = __builtin_amdgcn_wmma_f32_16x16x4_f32(false, a, false, b, (short)0, c, false, false);
  float s = c[0] + c[1] + c[2] + c[3] + c[4] + c[5] + c[6] + c[7];
  s += __shfl_xor(s, 16, 32);
  return s;
}
#endif // __HIP_DEVICE_COMPILE__

__global__ __launch_bounds__(256)
void edge_tile_kernel(const float* __restrict__ gen,
                      const float* __restrict__ real,
                      float* __restrict__ partial) {
#if defined(__HIP_DEVICE_COMPILE__)
  __shared__ __align__(16) float pr[N_PLANE];   // staged R plane, 38x40 (async dest)
  __shared__ __align__(16) float pg[N_PLANE];   // staged G plane
  __shared__ __align__(16) float pb[N_PLANE];   // staged B plane
  __shared__ float gry[N_GRAY];                 // gray w/ halo 3: 38x38
  __shared__ float hb[N_HB];                    // horizontal blur: 38 rows x 34 cols
  __shared__ float vb[N_VB];                    // full blur: 34x34 (zeroed outside image)
  __shared__ float wsum[8];

  const int tid  = threadIdx.x;
  const int ty   = blockIdx.x >> 4;        // 16x16 tiles of 32x32
  const int tx   = blockIdx.x & 15;
  const int oy0  = ty * TILE;
  const int ox0  = tx * TILE;
  const size_t imgOff = (size_t)blockIdx.y * (3u * HW);
  const bool fastx = (tx > 0) && (tx < 15); // cols [ox0-4, ox0+36) fully in-image

  const uint32_t prOff = (uint32_t)(uintptr_t)pr;
  const uint32_t pgOff = (uint32_t)(uintptr_t)pg;
  const uint32_t pbOff = (uint32_t)(uintptr_t)pb;

  float eg[4];
  float acc = 0.0f;

  for (int img = 0; img < 2; ++img) {
    const float* base = (img == 0 ? gen : real) + imgOff;
    const uint64_t sBase = (uint64_t)(uintptr_t)base;

    __syncthreads(); // protect LDS reuse across images

    // ---- Stage A: async-copy R,G,B 38x40 halo regions into LDS
    if (fastx) {
      // b128 fast path: 38 rows x 10 x (4 floats); only y needs clamping.
      #pragma unroll
      for (int k = 0; k < 2; ++k) {
        int p = tid + k * 256;
        if (p > 379) p = 379;                // duplicate tail: benign
        const int row = p / 10, c16 = p - row * 10;
        int gy = oy0 - 3 + row;  gy = gy < 0 ? 0 : (gy > IMG_H - 1 ? IMG_H - 1 : gy);
        const uint32_t vOff  = (uint32_t)((gy * IMG_W + (ox0 - 4)) * 4 + c16 * 16);
        const uint32_t lOff  = (uint32_t)(row * (GCOLS * 4) + c16 * 16);
        async_ld_b128(prOff + lOff, vOff,              sBase);
        async_ld_b128(pgOff + lOff, vOff + 4u * HW,    sBase);
        async_ld_b128(pbOff + lOff, vOff + 8u * HW,    sBase);
      }
    } else {
      // b32 clamped path for x-border tiles.
      #pragma unroll
      for (int k = 0; k < 6; ++k) {
        int p = tid + k * 256;
        if (p > N_PLANE - 1) p = N_PLANE - 1;
        const int iy = p / GCOLS, j = p - iy * GCOLS;
        int gy = oy0 - 3 + iy;  gy = gy < 0 ? 0 : (gy > IMG_H - 1 ? IMG_H - 1 : gy);
        int gx = ox0 - 4 + j;   gx = gx < 0 ? 0 : (gx > IMG_W - 1 ? IMG_W - 1 : gx);
        const uint32_t vOff = (uint32_t)((gy * IMG_W + gx) * 4);
        async_ld_b32(prOff + 4u * (uint32_t)p, vOff,            sBase);
        async_ld_b32(pgOff + 4u * (uint32_t)p, vOff + 4u * HW,  sBase);
        async_ld_b32(pbOff + 4u * (uint32_t)p, vOff + 8u * HW,  sBase);
      }
    }
    wait_async0();
    __syncthreads();

    // ---- Stage B: gray = 0.5*(0.299 r + 0.587 g + 0.114 b) + 0.5 ; 0 outside image
    // gray col ix in [0,38) is global gx = ox0-3+ix = plane col ix+1.
    #pragma unroll
    for (int k = 0; k < 6; ++k) {
      const int p = tid + k * 256;
      if (p < N_GRAY) {
        const int iy = p / GROWS, ix = p - iy * GROWS;
        const int gy = oy0 - 3 + iy, gx = ox0 - 3 + ix;
        const int q  = iy * GCOLS + ix + 1;
        float g = 0.0f;
        if (gy >= 0 && gy < IMG_H && gx >= 0 && gx < IMG_W)
          g = fmaf(0.299f, pr[q], fmaf(0.587f, pg[q], 0.114f * pb[q])) * 0.5f + 0.5f;
        gry[p] = g;
      }
    }
    __syncthreads();

    // ---- Stage C: horizontal 5-tap blur, 38 rows x 34 cols
    #pragma unroll
    for (int k = 0; k < 6; ++k) {
      const int p = tid + k * 256;
      if (p < N_HB) {
        const int y = p / HREG, x = p - y * HREG;
        const float* r = &gry[y * GROWS + x];
        hb[p] = fmaf(G0, r[0] + r[4], fmaf(G1, r[1] + r[3], G2 * r[2]));
      }
    }
    __syncthreads();

    // ---- Stage D: vertical 5-tap blur, 34x34; force 0 outside image (Laplacian zero-pad)
    #pragma unroll
    for (int k = 0; k < 5; ++k) {
      const int p = tid + k * 256;
      if (p < N_VB) {
        const int vy = p / HREG, x = p - vy * HREG;
        const float* cp = &hb[vy * HREG + x];
        float v = fmaf(G0, cp[0] + cp[4 * HREG], fmaf(G1, cp[HREG] + cp[3 * HREG], G2 * cp[2 * HREG]));
        const int gy = oy0 + vy - 1, gx = ox0 + x - 1;
        if (gy < 0 || gy >= IMG_H || gx < 0 || gx >= IMG_W) v = 0.0f;
        vb[p] = v;
      }
    }
    __syncthreads();

    // ---- Stage E: Laplacian + abs; 4 output pixels per thread
    #pragma unroll
    for (int k = 0; k < 4; ++k) {
      const int o  = tid + k * 256;
      const int oy = o >> 5, ox = o & 31;
      const int ci = (oy + 1) * HREG + ox + 1;
      const float e = fabsf(vb[ci - HREG] + vb[ci + HREG] + vb[ci - 1] + vb[ci + 1] - 4.0f * vb[ci]);
      if (img == 0) eg[k] = e;
      else          acc  += fabsf(eg[k] - e);
    }
  }

  __syncthreads();
  // ---- Block reduction: WMMA wave-sum, then 8 waves via LDS
  const float w = wave_reduce_wmma(acc);
  if ((tid & 31) == 0) wsum[tid >> 5] = w;
  __syncthreads();
  if (tid == 0) {
    float t = 0.0f;
    #pragma unroll
    for (int i = 0; i < 8; ++i) t += wsum[i];
    partial[blockIdx.y * 256 + blockIdx.x] = t;
  }
#endif
}

__global__ __launch_bounds__(256)
void final_reduce_kernel(const float* __restrict__ partial, float* __restrict__ out) {
#if defined(__HIP_DEVICE_COMPILE__)
  __shared__ float wsum[8];
  float acc = 0.0f;
  for (int i = threadIdx.x; i < 8192; i += 256) acc += partial[i];
  const float w = wave_reduce_wmma(acc);
  if ((threadIdx.x & 31) == 0) wsum[threadIdx.x >> 5] = w;
  __syncthreads();
  if (threadIdx.x == 0) {
    float t = 0.0f;
    #pragma unroll
    for (int i = 0; i < 8; ++i) t += wsum[i];
    out[0] = t * (1.0f / 8388608.0f);   // mean over 32*512*512
  }
#endif
}

extern "C" void kernel_launch(void* const* d_in, const int* in_sizes, int n_in,
                              void* d_out, int out_size, void* d_ws, size_t ws_size,
                              hipStream_t stream) {
  const float* gen  = (const float*)d_in[0];
  const float* real = (const float*)d_in[1];
  float* part = (float*)d_ws;           // 8192 floats = 32 KB of scratch
  (void)in_sizes; (void)n_in; (void)out_size; (void)ws_size;

  edge_tile_kernel<<<dim3(256, 32), 256, 0, stream>>>(gen, real, part);
  final_reduce_kernel<<<1, 256, 0, stream>>>(part, (float*)d_out);
}